// ASSRReconstructionLoss_20031727468905
// MI455X (gfx1250) — compile-verified
//
#include <hip/hip_runtime.h>
#include <cstdint>

typedef float        v2f   __attribute__((ext_vector_type(2)));
typedef float        v4f   __attribute__((ext_vector_type(4)));
typedef float        v8f   __attribute__((ext_vector_type(8)));
typedef unsigned int u32x4 __attribute__((ext_vector_type(4)));
typedef int          i32x4 __attribute__((ext_vector_type(4)));
typedef int          i32x8 __attribute__((ext_vector_type(8)));

#define BATCH   32
#define CHN     3
#define NIMG    (BATCH*CHN)          // 96
#define H       512
#define W       512
#define OH      128
#define OW      128
#define PIX_N   (NIMG*H*W)           // 25,165,824
#define LR_N    (NIMG*OH*OW)         // 1,572,864
#define NPIXBLK 2048

#define SLAB_W  64
#define NSLAB   (W/SLAB_W)           // 8
#define KCHUNK  32
#define NKCHUNK (H/KCHUNK)           // 16

// workspace layout (in floats)
#define WS_M_OFF   0
#define WS_M_SZ    (OH*W)            // 65536 floats (256 KB)
#define WS_PIX_OFF (WS_M_OFF + WS_M_SZ)
#define WS_LR_OFF  (WS_PIX_OFF + NPIXBLK)

// LDS layout (bytes) for the fused resize kernel
#define LDS_TMP_OFF      0
#define LDS_TMP_SZ       (OH*SLAB_W*4)        // 32768
#define LDS_STAGE_OFF    LDS_TMP_SZ           // 32768
#define LDS_STAGE_BUF_SZ (KCHUNK*SLAB_W*4)    // 8192
#define LDS_WRED_OFF     (LDS_STAGE_OFF + 2*LDS_STAGE_BUF_SZ)  // 49152
#define LDS_TOTAL        (LDS_WRED_OFF + 64)  // 49216 bytes

#define WMMA_F32X4(a,b,c) \
  __builtin_amdgcn_wmma_f32_16x16x4_f32(false,(a),false,(b),(short)0,(c),false,false)

// ---------------------------------------------------------------------------
// PyTorch bicubic kernel, a = -0.75
__device__ __forceinline__ float cubicw(float x) {
    const float a = -0.75f;
    float ax = fabsf(x), ax2 = ax*ax, ax3 = ax2*ax;
    float f1 = (a + 2.0f)*ax3 - (a + 3.0f)*ax2 + 1.0f;
    float f2 = a*ax3 - 5.0f*a*ax2 + 8.0f*a*ax - 4.0f*a;
    return (ax <= 1.0f) ? f1 : ((ax < 2.0f) ? f2 : 0.0f);
}

// Build the 128x512 antialiased-bicubic resize matrix (scale 4, align_corners
// =False, antialias, replicate border) — one block per output row.
__global__ void build_resize_matrix(float* __restrict__ M) {
    const int o = blockIdx.x;                 // output row, 0..127
    for (int i = threadIdx.x; i < W; i += blockDim.x) M[o*W + i] = 0.0f;
    __syncthreads();

    const float scale = 4.0f, s_aa = 4.0f;
    const int   ext   = 9;                    // ceil(2*s_aa)+1
    const int   span  = W + 2*ext;            // 530
    const float center = (o + 0.5f)*scale - 0.5f;

    float local = 0.0f;
    for (int t = threadIdx.x; t < span; t += blockDim.x) {
        int idx = t - ext;
        local += cubicw(((float)idx - center) / s_aa);
    }
    __shared__ float red[256];
    red[threadIdx.x] = local;
    __syncthreads();
    for (int s = 128; s > 0; s >>= 1) {
        if (threadIdx.x < s) red[threadIdx.x] += red[threadIdx.x + s];
        __syncthreads();
    }
    const float inv = 1.0f / red[0];

    for (int t = threadIdx.x; t < span; t += blockDim.x) {
        int idx = t - ext;
        float w = cubicw(((float)idx - center) / s_aa) * inv;
        int ic = min(max(idx, 0), W - 1);     // replicate border
        atomicAdd(&M[o*W + ic], w);
    }
}

// ---------------------------------------------------------------------------
// pix term: sum |pred - target| over 25.2M elements, float4 streaming.
// target is loaded NON-TEMPORAL so it does not evict pred_hr from the 192 MB
// L2: the fused resize kernel re-reads pred_hr right after this and should
// hit L2 instead of HBM (~100 MB of HBM traffic saved).
__global__ void pix_abs_reduce(const float* __restrict__ a,
                               const float* __restrict__ b,
                               float* __restrict__ partial) {
    const int n4 = PIX_N / 4;
    const v4f* __restrict__ a4 = (const v4f*)a;
    const v4f* __restrict__ b4 = (const v4f*)b;
    float s = 0.0f;
    for (int i = blockIdx.x*blockDim.x + threadIdx.x; i < n4;
         i += gridDim.x*blockDim.x) {
        v4f x = a4[i];                              // RT: keep pred in L2
        v4f y = __builtin_nontemporal_load(b4 + i); // NT: stream target
        s += fabsf(x.x - y.x) + fabsf(x.y - y.y)
           + fabsf(x.z - y.z) + fabsf(x.w - y.w);
    }
    __shared__ float red[256];
    red[threadIdx.x] = s;
    __syncthreads();
    for (int o = 128; o > 0; o >>= 1) {
        if (threadIdx.x < o) red[threadIdx.x] += red[threadIdx.x + o];
        __syncthreads();
    }
    if (threadIdx.x == 0) partial[blockIdx.x] = red[0];
}

// ---------------------------------------------------------------------------
// TDM: DMA a 32-row x 64-col f32 tile of the image into LDS.
__device__ __forceinline__ void tdm_load_chunk(const float* gsrc,
                                               unsigned lds_byte_off) {
    unsigned long long ga = (unsigned long long)(uintptr_t)gsrc;
    u32x4 g0;
    g0[0] = 1u;                                   // count=1, user descriptor
    g0[1] = lds_byte_off;                         // lds_addr
    g0[2] = (unsigned)(ga & 0xFFFFFFFFull);       // global_addr[31:0]
    g0[3] = (unsigned)((ga >> 32) & 0x1FFFFFFull) // global_addr[56:32]
          | (2u << 30);                           // type=2 ("image")
    i32x8 g1;
    g1[0] = 0x00020000;                           // data_size=2 -> 4 bytes
    g1[1] = (int)(512u << 16);                    // tensor_dim0 = 512 (lo16)
    g1[2] = (int)(512u << 16);                    // dim0 hi16=0 | tensor_dim1=512
    g1[3] = (int)((unsigned)SLAB_W << 16);        // dim1 hi16=0 | tile_dim0=64
    g1[4] = KCHUNK;                               // tile_dim1=32, tile_dim2=0
    g1[5] = 512;                                  // tensor_dim0_stride lo32
    g1[6] = 0;                                    // stride0 hi16 | stride1 lo16
    g1[7] = 0;                                    // stride1 hi32
    i32x4 z4 = {0, 0, 0, 0};
    i32x8 z8 = {0, 0, 0, 0, 0, 0, 0, 0};
    __builtin_amdgcn_tensor_load_to_lds(g0, g1, z4, z4, z8, 0);
}

// ---------------------------------------------------------------------------
// Fused per-image kernel: pred_lr = M * img * M^T via f32 WMMA, tmp kept in
// LDS (slabbed), TDM-staged inputs, fused |pred_lr - lr_ref| reduction.
// grid = 96 blocks, 512 threads = 16 waves.
__global__ __launch_bounds__(512)
void fused_resize_lr(const float* __restrict__ pred_hr,
                     const float* __restrict__ Mg,
                     const float* __restrict__ lr_ref,
                     float* __restrict__ lr_partial) {
    __shared__ __align__(16) unsigned char lds_raw[LDS_TOTAL];
    float* tmp    = (float*)(lds_raw + LDS_TMP_OFF);    // [128][64]
    float* stageF = (float*)(lds_raw + LDS_STAGE_OFF);  // 2 x [32][64]
    float* wred   = (float*)(lds_raw + LDS_WRED_OFF);   // [16]

    const int lane  = threadIdx.x & 31;
    const int wid   = threadIdx.x >> 5;
    const int lrow  = lane & 15;          // 0..15
    const int khalf = (lane >> 4) << 1;   // 0 or 2
    const int rbase = (lane >> 4) << 3;   // 0 or 8 (C/D row offset)

    const float* img = pred_hr + (size_t)blockIdx.x * H * W;

    // GEMM1 tile ownership: tmp is 8x4 tiles of 16x16; wave owns 2 tiles.
    const int t1  = wid * 2;
    const int to1 = t1 >> 2;              // tile row (shared by both tiles)
    const int tw1 = t1 & 3;               // tile cols tw1, tw1+1

    // GEMM2 tile ownership: out is 8x8 tiles; wave owns 4 tiles (same row).
    const int t2  = wid * 4;
    const int to2 = t2 >> 3;
    const int tp2 = t2 & 7;

    const float* Arow1 = Mg + (size_t)(to1*16 + lrow) * W;   // GEMM1 A row

    v8f oacc[4];
    #pragma unroll
    for (int j = 0; j < 4; ++j)
        oacc[j] = (v8f){0.f,0.f,0.f,0.f,0.f,0.f,0.f,0.f};

    for (int s = 0; s < NSLAB; ++s) {
        const int w0 = s * SLAB_W;

        // ---- Phase A: tmp[:, w0:w0+64] = M @ img[:, w0:w0+64] -------------
        v8f acc0 = (v8f){0.f,0.f,0.f,0.f,0.f,0.f,0.f,0.f};
        v8f acc1 = acc0;

        if (wid == 0)  // prologue DMA: chunk 0 -> buffer 0
            tdm_load_chunk(img + w0, LDS_STAGE_OFF);

        for (int kc = 0; kc < NKCHUNK; ++kc) {
            const int buf = kc & 1;
            if (wid == 0) {
                if (kc + 1 < NKCHUNK) {
                    tdm_load_chunk(img + (size_t)(kc + 1) * KCHUNK * W + w0,
                                   LDS_STAGE_OFF + (buf ^ 1) * LDS_STAGE_BUF_SZ);
                    __builtin_amdgcn_s_wait_tensorcnt(1);
                } else {
                    __builtin_amdgcn_s_wait_tensorcnt(0);
                }
            }
            __syncthreads();  // stage[buf] now holds rows kc*32..kc*32+31

            const float* S = stageF + buf * (KCHUNK * SLAB_W);
            const int kc0  = kc * KCHUNK;

            // Preload the whole 32-deep k-chunk of fragments so the loads
            // clause and the 16 WMMAs issue back-to-back without per-step
            // loadcnt stalls.
            v2f afr[8], bfr0[8], bfr1[8];
            #pragma unroll
            for (int kk = 0; kk < KCHUNK / 4; ++kk) {
                const int k = kk * 4;
                afr[kk].x  = Arow1[kc0 + k + khalf];
                afr[kk].y  = Arow1[kc0 + k + khalf + 1];
                bfr0[kk].x = S[(k + khalf    ) * SLAB_W + tw1*16     + lrow];
                bfr0[kk].y = S[(k + khalf + 1) * SLAB_W + tw1*16     + lrow];
                bfr1[kk].x = S[(k + khalf    ) * SLAB_W + (tw1+1)*16 + lrow];
                bfr1[kk].y = S[(k + khalf + 1) * SLAB_W + (tw1+1)*16 + lrow];
            }
            #pragma unroll
            for (int kk = 0; kk < KCHUNK / 4; ++kk) {
                acc0 = WMMA_F32X4(afr[kk], bfr0[kk], acc0);
                acc1 = WMMA_F32X4(afr[kk], bfr1[kk], acc1);
            }
            __syncthreads();  // everyone done with stage[buf] before refill
        }

        // spill Phase A accumulators into LDS tmp
        {
            union { v8f v; float f[8]; } u0, u1;
            u0.v = acc0; u1.v = acc1;
            #pragma unroll
            for (int r = 0; r < 8; ++r) {
                const int row = to1*16 + r + rbase;
                tmp[row*SLAB_W + tw1*16     + lrow] = u0.f[r];
                tmp[row*SLAB_W + (tw1+1)*16 + lrow] = u1.f[r];
            }
        }
        __syncthreads();

        // ---- Phase B: out += tmp_slab @ (M[:, w0:w0+64])^T ----------------
        // Grouped preload (4 k-steps at a time) to overlap LDS/global loads
        // with the WMMA stream.
        #pragma unroll
        for (int kb = 0; kb < SLAB_W; kb += 16) {
            v2f aB[4];
            v2f bB[4][4];
            #pragma unroll
            for (int g = 0; g < 4; ++g) {
                const int k = kb + g*4;
                aB[g].x = tmp[(to2*16 + lrow)*SLAB_W + k + khalf];
                aB[g].y = tmp[(to2*16 + lrow)*SLAB_W + k + khalf + 1];
                #pragma unroll
                for (int j = 0; j < 4; ++j) {
                    const float* Brow = Mg + (size_t)((tp2 + j)*16 + lrow) * W;
                    bB[g][j].x = Brow[w0 + k + khalf];
                    bB[g][j].y = Brow[w0 + k + khalf + 1];
                }
            }
            #pragma unroll
            for (int g = 0; g < 4; ++g)
                #pragma unroll
                for (int j = 0; j < 4; ++j)
                    oacc[j] = WMMA_F32X4(aB[g], bB[g][j], oacc[j]);
        }
        __syncthreads();  // protect tmp before next slab overwrites it
    }

    // ---- fused |pred_lr - lr_ref| reduction (pred_lr never hits memory) ---
    const float* lrimg = lr_ref + (size_t)blockIdx.x * OH * OW;
    float local = 0.0f;
    #pragma unroll
    for (int j = 0; j < 4; ++j) {
        union { v8f v; float f[8]; } u;
        u.v = oacc[j];
        #pragma unroll
        for (int r = 0; r < 8; ++r) {
            const int row = to2*16 + r + rbase;
            const int col = (tp2 + j)*16 + lrow;
            local += fabsf(u.f[r] - lrimg[row*OW + col]);
        }
    }
    #pragma unroll
    for (int off = 16; off > 0; off >>= 1)
        local += __shfl_xor(local, off, 32);
    if (lane == 0) wred[wid] = local;
    __syncthreads();
    if (threadIdx.x == 0) {
        float t = 0.0f;
        #pragma unroll
        for (int i = 0; i < 16; ++i) t += wred[i];
        lr_partial[blockIdx.x] = t;
    }
}

// ---------------------------------------------------------------------------
__global__ void finalize_loss(const float* __restrict__ pixp,
                              const float* __restrict__ lrp,
                              float* __restrict__ out) {
    __shared__ float red[256];
    float s = 0.0f;
    for (int i = threadIdx.x; i < NPIXBLK; i += 256) s += pixp[i];
    red[threadIdx.x] = s;
    __syncthreads();
    for (int o = 128; o > 0; o >>= 1) {
        if (threadIdx.x < o) red[threadIdx.x] += red[threadIdx.x + o];
        __syncthreads();
    }
    const float pixsum = red[0];
    __syncthreads();

    float t = 0.0f;
    for (int i = threadIdx.x; i < NIMG; i += 256) t += lrp[i];
    red[threadIdx.x] = t;
    __syncthreads();
    for (int o = 128; o > 0; o >>= 1) {
        if (threadIdx.x < o) red[threadIdx.x] += red[threadIdx.x + o];
        __syncthreads();
    }
    const float lrsum = red[0];

    if (threadIdx.x == 0) {
        const float pix       = pixsum / (float)PIX_N;
        const float lr_term   = lrsum  / (float)LR_N;
        const float pair_term = 0.0f;
        const float consist   = 1.0f * lr_term + 1.0f * pair_term; // LAM_LR, LAM_PAIR
        const float total     = pix + 0.1f * consist;              // LAM_CONSIST
        out[0] = total;
        out[1] = pix;
        out[2] = consist;
        out[3] = lr_term;
        out[4] = pair_term;
    }
}

// ---------------------------------------------------------------------------
extern "C" void kernel_launch(void* const* d_in, const int* in_sizes, int n_in,
                              void* d_out, int out_size, void* d_ws, size_t ws_size,
                              hipStream_t stream) {
    const float* pred_hr   = (const float*)d_in[0];
    const float* target_hr = (const float*)d_in[1];
    const float* lr_ref    = (const float*)d_in[2];
    // d_in[3] = scale (uniform 4.0) -> sizes are static, unused on device.

    float* out  = (float*)d_out;
    float* wsf  = (float*)d_ws;
    float* M    = wsf + WS_M_OFF;
    float* pixp = wsf + WS_PIX_OFF;
    float* lrp  = wsf + WS_LR_OFF;

    build_resize_matrix<<<OH, 256, 0, stream>>>(M);
    pix_abs_reduce<<<NPIXBLK, 256, 0, stream>>>(pred_hr, target_hr, pixp);
    fused_resize_lr<<<NIMG, 512, 0, stream>>>(pred_hr, M, lr_ref, lrp);
    finalize_loss<<<1, 256, 0, stream>>>(pixp, lrp, out);
}